// QCMHA_29652454211746
// MI455X (gfx1250) — compile-verified
//
#include <hip/hip_runtime.h>
#include <math.h>

// Problem constants (from reference)
#define Tt   384
#define Dd   256
#define Hh   8
#define DHd  32
#define BT   768        // B * T

// Attention chunking
#define CH    24        // s-rows staged per chunk (384 = 16 * 24)
#define PITCH 260       // LDS row pitch in floats: 260 % 64 = 4 -> <=2-way bank
                        // conflict for lane-strided reads, 16B aligned rows

typedef __attribute__((ext_vector_type(2))) float v2f;
typedef __attribute__((ext_vector_type(8))) float v8f;

// Pointer-to-16B-vector types in explicit address spaces for the async builtin
typedef int v4i __attribute__((vector_size(16)));
typedef __attribute__((address_space(1))) v4i* gptr16;   // global
typedef __attribute__((address_space(3))) v4i* lptr16;   // LDS

// ---- gfx1250 async global->LDS copy (ASYNCcnt path), with safe fallback ----
__device__ __forceinline__ void async_copy16(const float* g, float* l) {
#if __has_builtin(__builtin_amdgcn_global_load_async_to_lds_b128)
  __builtin_amdgcn_global_load_async_to_lds_b128(
      (gptr16)(uintptr_t)g, (lptr16)l, 0, 0);
#else
  float4 t = *(const float4*)g;
  *(float4*)l = t;
#endif
}

__device__ __forceinline__ void wait_async() {
#if __has_builtin(__builtin_amdgcn_s_wait_asynccnt)
  __builtin_amdgcn_s_wait_asynccnt(0);
#else
  asm volatile("s_wait_asynccnt 0" ::: "memory");
#endif
}

// -------------------------------------------------------------------------
// Kernel 1: QKV projection.  Y = X @ W_in^T + b_in scattered into head-major
// Q/K/V buffers (row = b*T+t, col = h*32+d); Q pre-scaled by 1/sqrt(dh).
// M=768, N=768, K=256.  16x16 tile per wave via V_WMMA_F32_16X16X4_F32.
// -------------------------------------------------------------------------
__global__ __launch_bounds__(128) void qkv_wmma(const float* __restrict__ X,
                                                const float* __restrict__ W,
                                                const float* __restrict__ bias,
                                                float* __restrict__ Qb,
                                                float* __restrict__ Kb,
                                                float* __restrict__ Vb) {
  const int tile = blockIdx.x * 4 + (threadIdx.x >> 5);   // 2304 tiles
  const int tm   = tile / 48;
  const int tn   = tile % 48;
  const int lane = threadIdx.x & 31;
  const int half = lane >> 4;
  const int l    = lane & 15;

  const float* xrow = X + (size_t)(tm * 16 + l) * Dd + 2 * half;
  const float* wrow = W + (size_t)(tn * 16 + l) * Dd + 2 * half;

  v8f acc = {};
#pragma unroll 8
  for (int k0 = 0; k0 < Dd; k0 += 4) {
    v2f a, b;
    a.x = xrow[k0]; a.y = xrow[k0 + 1];
    b.x = wrow[k0]; b.y = wrow[k0 + 1];
    acc = __builtin_amdgcn_wmma_f32_16x16x4_f32(false, a, false, b,
                                                (short)0, acc, false, false);
  }

  const int   n    = tn * 16 + l;
  const int   h    = n / 96;
  const int   j    = n % 96;
  float*      dst  = (j < 32) ? Qb : (j < 64) ? Kb : Vb;
  const int   dcol = h * DHd + (j & 31);
  const float bv   = bias[n];
  const float scl  = (j < 32) ? 0.17677669529663687f : 1.0f;  // 1/sqrt(32)
#pragma unroll
  for (int r = 0; r < 8; ++r) {
    const int m = tm * 16 + r + 8 * half;
    dst[(size_t)m * Dd + dcol] = (acc[r] + bv) * scl;
  }
}

// -------------------------------------------------------------------------
// Kernel 2: SK[b,h,t,s] = q_scaled[b,t,h,:] . k[b,s,h,:]  -  mask[b,t,s]*1e9
// Batched WMMA GEMM: 16 batches (b*8+h), M=N=384, K=32.
// -------------------------------------------------------------------------
__global__ __launch_bounds__(128) void qk_wmma(const float* __restrict__ Qb,
                                               const float* __restrict__ Kb,
                                               const float* __restrict__ mask,
                                               float* __restrict__ SK) {
  const int gtile = blockIdx.x * 4 + (threadIdx.x >> 5);  // 16*576 tiles
  const int batch = gtile / 576;                          // b*8 + h
  const int tile  = gtile % 576;
  const int tm    = tile / 24;
  const int tn    = tile % 24;
  const int b     = batch >> 3;
  const int h     = batch & 7;
  const int lane  = threadIdx.x & 31;
  const int half  = lane >> 4;
  const int l     = lane & 15;

  const float* arow = Qb + (size_t)(b * Tt + tm * 16 + l) * Dd + h * DHd + 2 * half;
  const float* brow = Kb + (size_t)(b * Tt + tn * 16 + l) * Dd + h * DHd + 2 * half;

  v8f acc = {};
#pragma unroll
  for (int k0 = 0; k0 < DHd; k0 += 4) {
    v2f a, bb;
    a.x  = arow[k0]; a.y  = arow[k0 + 1];
    bb.x = brow[k0]; bb.y = brow[k0 + 1];
    acc = __builtin_amdgcn_wmma_f32_16x16x4_f32(false, a, false, bb,
                                                (short)0, acc, false, false);
  }

  const int s = tn * 16 + l;
#pragma unroll
  for (int r = 0; r < 8; ++r) {
    const int t = tm * 16 + r + 8 * half;
    SK[((size_t)batch * Tt + t) * Tt + s] =
        acc[r] - mask[((size_t)(b * Tt + t)) * Tt + s] * 1.0e9f;
  }
}

// -------------------------------------------------------------------------
// Kernel 3: fused attention.  One block per (b,t), one wave per head.
// Per 24-row chunk: async-stage rel & V into LDS; phase A (lane = s) adds
// q.rel to the precomputed SK scores; one online-softmax step per chunk;
// phase B (lane = d) rank-24 accumulate of p*(V+rel) from the same LDS data.
// rel is read from HBM exactly once.
// -------------------------------------------------------------------------
__global__ __launch_bounds__(256) void attn_fused(const float* __restrict__ Qb,
                                                  const float* __restrict__ Vb,
                                                  const float* __restrict__ rel,
                                                  const float* __restrict__ SK,
                                                  float* __restrict__ AO) {
  __shared__ float Rl[CH * PITCH];   // 24 x 260 floats = 24.96 KB
  __shared__ float Vl[CH * PITCH];

  const int bt   = blockIdx.x;
  const int b    = bt / Tt;
  const int t    = bt - b * Tt;
  const int tid  = threadIdx.x;
  const int wave = tid >> 5;          // head
  const int lane = tid & 31;
  const int hd   = wave * DHd;

  // broadcast this head's q (already 1/sqrt(dh)-scaled) into a register array
  const float qv = Qb[(size_t)bt * Dd + hd + lane];
  float qreg[DHd];
#pragma unroll
  for (int d = 0; d < DHd; ++d) qreg[d] = __shfl(qv, d, 32);

  const float* SKrow   = SK  + ((size_t)(b * Hh + wave) * Tt + t) * Tt;
  const float* relbase = rel + (size_t)bt * Tt * Dd;
  const float* vbase   = Vb  + (size_t)(b * Tt) * Dd;

  float m = -INFINITY, lsum = 0.0f, acc = 0.0f;

  for (int s0 = 0; s0 < Tt; s0 += CH) {
    // ---- stage chunk: 24 x 1KB of rel and V, contiguous, async ----
    const float* gR = relbase + (size_t)s0 * Dd;   // 24 KB contiguous
    const float* gV = vbase   + (size_t)s0 * Dd;
#pragma unroll
    for (int u = 0; u < 6; ++u) {
      const int f   = tid + (u << 8);              // float4 index 0..1535
      const int row = f >> 6;
      const int c4  = (f & 63) * 4;
      async_copy16(gR + (size_t)f * 4, Rl + row * PITCH + c4);
      async_copy16(gV + (size_t)f * 4, Vl + row * PITCH + c4);
    }
    wait_async();
    __syncthreads();

    // ---- phase A: lane = s-in-chunk; score = SK + q.rel ----
    float sc = -INFINITY;
    if (lane < CH) {
      sc = SKrow[s0 + lane];
      const float* rj = Rl + lane * PITCH + hd;
#pragma unroll
      for (int d = 0; d < DHd; d += 4) {
        const float4 rv = *(const float4*)(rj + d);
        sc += qreg[d] * rv.x + qreg[d + 1] * rv.y +
              qreg[d + 2] * rv.z + qreg[d + 3] * rv.w;
      }
    }

    // ---- one online-softmax step per 24 scores ----
    float cm = sc;
#pragma unroll
    for (int off = 16; off; off >>= 1) cm = fmaxf(cm, __shfl_xor(cm, off, 32));
    const float mn   = fmaxf(m, cm);
    const float corr = __expf(m - mn);            // exp(-inf)=0 on first chunk
    const float p    = __expf(sc - mn);           // 0 for idle lanes (sc=-inf)
    float ps = p;
#pragma unroll
    for (int off = 16; off; off >>= 1) ps += __shfl_xor(ps, off, 32);
    lsum = lsum * corr + ps;
    acc *= corr;
    m = mn;

    // ---- phase B: lane = d; acc += p_j * (V + rel) ----
#pragma unroll 8
    for (int j = 0; j < CH; ++j) {
      const float pj = __shfl(p, j, 32);
      acc = fmaf(pj, Vl[j * PITCH + hd + lane] + Rl[j * PITCH + hd + lane], acc);
    }
    __syncthreads();   // protect LDS before next chunk's staging
  }

  AO[(size_t)bt * Dd + hd + lane] = acc / lsum;
}

// -------------------------------------------------------------------------
// Kernel 4: output projection.  out = AO @ W_out^T + b_out.  M=768,N=256,K=256.
// -------------------------------------------------------------------------
__global__ __launch_bounds__(128) void proj_wmma(const float* __restrict__ X,
                                                 const float* __restrict__ W,
                                                 const float* __restrict__ bias,
                                                 float* __restrict__ Y) {
  const int tile = blockIdx.x * 4 + (threadIdx.x >> 5);   // 768 tiles
  const int tm   = tile / 16;
  const int tn   = tile % 16;
  const int lane = threadIdx.x & 31;
  const int half = lane >> 4;
  const int l    = lane & 15;

  const float* xrow = X + (size_t)(tm * 16 + l) * Dd + 2 * half;
  const float* wrow = W + (size_t)(tn * 16 + l) * Dd + 2 * half;

  v8f acc = {};
#pragma unroll 8
  for (int k0 = 0; k0 < Dd; k0 += 4) {
    v2f a, b;
    a.x = xrow[k0]; a.y = xrow[k0 + 1];
    b.x = wrow[k0]; b.y = wrow[k0 + 1];
    acc = __builtin_amdgcn_wmma_f32_16x16x4_f32(false, a, false, b,
                                                (short)0, acc, false, false);
  }

  const int   n  = tn * 16 + l;
  const float bv = bias[n];
#pragma unroll
  for (int r = 0; r < 8; ++r) {
    const int m = tm * 16 + r + 8 * half;
    Y[(size_t)m * Dd + n] = acc[r] + bv;
  }
}

// -------------------------------------------------------------------------
extern "C" void kernel_launch(void* const* d_in, const int* in_sizes, int n_in,
                              void* d_out, int out_size, void* d_ws, size_t ws_size,
                              hipStream_t stream) {
  (void)in_sizes; (void)n_in; (void)out_size; (void)ws_size;

  const float* query = (const float*)d_in[0];   // (2,384,256)
  const float* rel   = (const float*)d_in[1];   // (2,384,384,256)
  const float* mask  = (const float*)d_in[2];   // (2,384,384)
  const float* W_in  = (const float*)d_in[3];   // (768,256)
  const float* b_in  = (const float*)d_in[4];   // (768)
  const float* W_out = (const float*)d_in[5];   // (256,256)
  const float* b_out = (const float*)d_in[6];   // (256)
  float*       out   = (float*)d_out;           // (2,384,256)

  // Workspace: Q,K,V,AO (768x256 each) + SK (16x384x384)  ~= 12.6 MB
  float* Qb = (float*)d_ws;
  float* Kb = Qb + (size_t)BT * Dd;
  float* Vb = Kb + (size_t)BT * Dd;
  float* AO = Vb + (size_t)BT * Dd;
  float* SK = AO + (size_t)BT * Dd;

  qkv_wmma  <<< 576, 128, 0, stream>>>(query, W_in, b_in, Qb, Kb, Vb);
  qk_wmma   <<<2304, 128, 0, stream>>>(Qb, Kb, mask, SK);
  attn_fused<<<BT,   256, 0, stream>>>(Qb, Vb, rel, SK, AO);
  proj_wmma <<< 192, 128, 0, stream>>>(AO, W_out, b_out, out);
}